// IGCCF_60318520705224
// MI455X (gfx1250) — compile-verified
//
#include <hip/hip_runtime.h>
#include <stdint.h>

// Problem constants (match the reference)
#define I_CNT     100000
#define U_CNT     131072
#define E_DIM     64
#define NNZ_S_CNT 1500000
#define NNZ_U_CNT 4194304

#define WAVE  32
#define WPB   8          // waves per block (256 threads = 8 wave32)
#define CHUNK 32         // nnz staged per wave per iteration
#define ROWS_PER_WAVE 16 // one 16-row WMMA tile per wave

typedef __attribute__((ext_vector_type(2))) float v2f;
typedef __attribute__((ext_vector_type(8))) float v8f;

// ---------------------------------------------------------------------------
// JAX threefry2x32 (Random123), exactly as jax.random uses it.
// ---------------------------------------------------------------------------
__device__ __forceinline__ uint32_t rotl32(uint32_t x, int d) {
    return (x << d) | (x >> (32 - d));
}

__device__ __forceinline__ void threefry2x32(uint32_t k0, uint32_t k1,
                                             uint32_t& x0, uint32_t& x1) {
    const uint32_t k2 = k0 ^ k1 ^ 0x1BD11BDAu;
    x0 += k0; x1 += k1;
#define TF_R4(a, b, c, d)                                        \
    x0 += x1; x1 = rotl32(x1, a); x1 ^= x0;                      \
    x0 += x1; x1 = rotl32(x1, b); x1 ^= x0;                      \
    x0 += x1; x1 = rotl32(x1, c); x1 ^= x0;                      \
    x0 += x1; x1 = rotl32(x1, d); x1 ^= x0;
    TF_R4(13, 15, 26, 6);  x0 += k1; x1 += k2 + 1u;
    TF_R4(17, 29, 16, 24); x0 += k2; x1 += k0 + 2u;
    TF_R4(13, 15, 26, 6);  x0 += k0; x1 += k1 + 3u;
    TF_R4(17, 29, 16, 24); x0 += k1; x1 += k2 + 4u;
    TF_R4(13, 15, 26, 6);  x0 += k2; x1 += k0 + 5u;
#undef TF_R4
}

// jax.random.bernoulli(jax.random.key(42), 0.8, (NNZ_U,)) -> keep ? 1/0.8 : 0
__device__ __forceinline__ float dropout_scale(uint32_t j) {
    const uint32_t half = (uint32_t)(NNZ_U_CNT / 2);
    const uint32_t i  = (j < half) ? j : (j - half);
    uint32_t x0 = i, x1 = i + half;
    threefry2x32(0u, 42u, x0, x1);
    const uint32_t bits = (j < half) ? x0 : x1;
    const float u = __uint_as_float((bits >> 9) | 0x3f800000u) - 1.0f;
    return (u < 0.8f) ? 1.25f : 0.0f;
}

// ---------------------------------------------------------------------------
// Indicator-routed COO SpMM on the f32 matrix pipe.
// One wave32 owns 16 consecutive output rows. For each group of 4 nnz:
//   A[m][k] = (rows[k]==base+m) ? vals[k] : 0     (16x4 routing matrix, fp32)
//   B[k][n] = x[cols[k]][16t+n]                    (4 gathered rows, tile t)
//   D_t    += A x B   via V_WMMA_F32_16X16X4_F32   (t = 0..3 covers E=64)
// Chunks of 32 nnz are staged into LDS (cols/rows via CDNA5 async global->LDS,
// ASYNCcnt); lanes beyond the chunk sanitize their slots to {0,-1,0} so the
// hot loop needs only unconditional ds_load_b64 pair reads — no divergent
// branches, no bpermute chains. EXEC is all-ones at every WMMA.
// ---------------------------------------------------------------------------
template <bool DROPOUT>
__global__ __launch_bounds__(WAVE * WPB)
void spmm_wmma_tile(const float* __restrict__ x,
                    const float* __restrict__ vals,
                    const int*   __restrict__ rows,
                    const int*   __restrict__ cols,
                    float*       __restrict__ out,
                    int nnz, int n_rows) {
    __shared__ int   lds_cols[WPB * CHUNK];
    __shared__ int   lds_rows[WPB * CHUNK];
    __shared__ float lds_vals[WPB * CHUNK];

    const int lane = threadIdx.x & (WAVE - 1);
    const int wid  = threadIdx.x >> 5;
    const int tile = blockIdx.x * WPB + wid;
    const int base = tile * ROWS_PER_WAVE;
    if (base >= n_rows) return;

    // nnz range covering rows [base, base+16) — wave-uniform binary searches.
    int lo = 0, hi = nnz;
    while (lo < hi) { int mid = (lo + hi) >> 1; if (rows[mid] < base) lo = mid + 1; else hi = mid; }
    const int start = lo;
    hi = nnz;
    while (lo < hi) { int mid = (lo + hi) >> 1; if (rows[mid] < base + ROWS_PER_WAVE) lo = mid + 1; else hi = mid; }
    const int end = lo;

    const int h        = lane >> 4;   // lane half: selects nnz pair {2h, 2h+1}
    const int n        = lane & 15;   // tile column for B/D; row m for A
    const int rowmatch = base + n;
    const int wbase    = wid * CHUNK;
    const uint32_t lds_cols_base = (uint32_t)(uintptr_t)(&lds_cols[wbase]);
    const uint32_t lds_rows_base = (uint32_t)(uintptr_t)(&lds_rows[wbase]);
    const uint32_t lds_vals_base = (uint32_t)(uintptr_t)(&lds_vals[wbase]);

    v8f acc0 = {}, acc1 = {}, acc2 = {}, acc3 = {};

    for (int cs = start; cs < end; cs += CHUNK) {
        const int cnt = min(CHUNK, end - cs);
        if (lane < cnt) {
            const int j = cs + lane;
            const uint32_t off = (uint32_t)lane * 4u;
            const uint64_t gc = (uint64_t)(uintptr_t)(cols + j);
            const uint64_t gr = (uint64_t)(uintptr_t)(rows + j);
            // CDNA5 async global->LDS staging of the chunk (ASYNCcnt-tracked).
            asm volatile("global_load_async_to_lds_b32 %0, %1, off"
                         :: "v"(lds_cols_base + off), "v"(gc) : "memory");
            asm volatile("global_load_async_to_lds_b32 %0, %1, off"
                         :: "v"(lds_rows_base + off), "v"(gr) : "memory");
            if (DROPOUT) {
                // scaled once per nnz here; same-wave DS ops stay in order
                lds_vals[wbase + lane] = vals[j] * dropout_scale((uint32_t)j);
            } else {
                const uint64_t gv = (uint64_t)(uintptr_t)(vals + j);
                asm volatile("global_load_async_to_lds_b32 %0, %1, off"
                             :: "v"(lds_vals_base + off), "v"(gv) : "memory");
            }
        } else {
            // sanitize padded slots: contribute nothing, gather row 0 safely
            lds_cols[wbase + lane] = 0;
            lds_rows[wbase + lane] = -1;
            lds_vals[wbase + lane] = 0.0f;
        }
        asm volatile("s_wait_asynccnt 0x0" ::: "memory");

        if (lane < cnt) {
            const int cmine = lds_cols[wbase + lane];
            const float* prow = x + (size_t)cmine * E_DIM;
            __builtin_prefetch(prow, 0, 1);
            __builtin_prefetch(prow + 32, 0, 1);
        }

        for (int k = 0; k < cnt; k += 4) {            // wave-uniform trip count
            const int p = wbase + k + 2 * h;          // even -> 8B aligned
            const int2   cpair = *(const int2*)  (&lds_cols[p]);
            const int2   rpair = *(const int2*)  (&lds_rows[p]);
            const float2 vpair = *(const float2*)(&lds_vals[p]);

            v2f a;                                    // A[m][2h], A[m][2h+1]
            a.x = (rpair.x == rowmatch) ? vpair.x : 0.0f;
            a.y = (rpair.y == rowmatch) ? vpair.y : 0.0f;

            const float* p0 = x + (size_t)cpair.x * E_DIM + n;
            const float* p1 = x + (size_t)cpair.y * E_DIM + n;
            v2f b0, b1, b2, b3;                       // B rows K=2h (x), K=2h+1 (y)
            b0.x = p0[0];  b0.y = p1[0];
            b1.x = p0[16]; b1.y = p1[16];
            b2.x = p0[32]; b2.y = p1[32];
            b3.x = p0[48]; b3.y = p1[48];

            acc0 = __builtin_amdgcn_wmma_f32_16x16x4_f32(false, a, false, b0,
                                                         (short)0, acc0, false, false);
            acc1 = __builtin_amdgcn_wmma_f32_16x16x4_f32(false, a, false, b1,
                                                         (short)0, acc1, false, false);
            acc2 = __builtin_amdgcn_wmma_f32_16x16x4_f32(false, a, false, b2,
                                                         (short)0, acc2, false, false);
            acc3 = __builtin_amdgcn_wmma_f32_16x16x4_f32(false, a, false, b3,
                                                         (short)0, acc3, false, false);
        }
    }

    // D layout: lanes 0-15 hold rows base+0..7 (VGPR g = row g), lanes 16-31
    // hold rows base+8..15; column n within each 16-wide tile.
    float* o = out + (size_t)(base + h * 8) * E_DIM + n;
#pragma unroll
    for (int g = 0; g < 8; ++g) {
        o[(size_t)g * E_DIM +  0] = acc0[g];
        o[(size_t)g * E_DIM + 16] = acc1[g];
        o[(size_t)g * E_DIM + 32] = acc2[g];
        o[(size_t)g * E_DIM + 48] = acc3[g];
    }
}

// ---------------------------------------------------------------------------
// d_in order (setup_inputs): item_emb, S_vals, S_rows, S_cols,
//                            urm_vals, urm_rows, urm_cols
// d_out: [user_emb (U*E) | x (I*E)]  (reference return order)
// d_ws : two I*E float ping-pong buffers (51.2 MB)
// ---------------------------------------------------------------------------
extern "C" void kernel_launch(void* const* d_in, const int* in_sizes, int n_in,
                              void* d_out, int out_size, void* d_ws, size_t ws_size,
                              hipStream_t stream) {
    const float* item_emb = (const float*)d_in[0];
    const float* S_vals   = (const float*)d_in[1];
    const int*   S_rows   = (const int*)  d_in[2];
    const int*   S_cols   = (const int*)  d_in[3];
    const float* urm_vals = (const float*)d_in[4];
    const int*   urm_rows = (const int*)  d_in[5];
    const int*   urm_cols = (const int*)  d_in[6];

    float* user_out = (float*)d_out;                               // U*E
    float* x_out    = (float*)d_out + (size_t)U_CNT * E_DIM;       // I*E
    float* bufA     = (float*)d_ws;
    float* bufB     = bufA + (size_t)I_CNT * E_DIM;

    const dim3 blk(WAVE * WPB);
    const int tilesI = (I_CNT + ROWS_PER_WAVE - 1) / ROWS_PER_WAVE;  // 6250
    const int tilesU = (U_CNT + ROWS_PER_WAVE - 1) / ROWS_PER_WAVE;  // 8192
    const int gridI  = (tilesI + WPB - 1) / WPB;
    const int gridU  = (tilesU + WPB - 1) / WPB;

    // x <- S @ x, three hops (ping-pong, last hop lands in d_out's x slice)
    spmm_wmma_tile<false><<<gridI, blk, 0, stream>>>(item_emb, S_vals, S_rows, S_cols,
                                                     bufA, NNZ_S_CNT, I_CNT);
    spmm_wmma_tile<false><<<gridI, blk, 0, stream>>>(bufA, S_vals, S_rows, S_cols,
                                                     bufB, NNZ_S_CNT, I_CNT);
    spmm_wmma_tile<false><<<gridI, blk, 0, stream>>>(bufB, S_vals, S_rows, S_cols,
                                                     x_out, NNZ_S_CNT, I_CNT);
    // user_emb <- dropout(URM) @ x
    spmm_wmma_tile<true><<<gridU, blk, 0, stream>>>(x_out, urm_vals, urm_rows, urm_cols,
                                                    user_out, NNZ_U_CNT, U_CNT);
}